// MLPs_71880572666072
// MI455X (gfx1250) — compile-verified
//
#include <hip/hip_runtime.h>

typedef __attribute__((ext_vector_type(2))) float v2f;
typedef __attribute__((ext_vector_type(8))) float v8f;

#define VOCAB   50000
#define EMB     300
#define HID     256
#define OUTD    3
#define NASP    8
#define BATCH   8192
#define SEQ     128
#define ROWS    64      // samples per tile in the MLP kernel
#define RT      4       // 16-row WMMA tiles per sample tile (ROWS/16)

// ---------------------------------------------------------------------------
// Kernel 1: embedding gather + mean pool.  x[s][d] = mean_t emb[ids[s][t]][d]
// emb (60MB) is L2-resident on MI455X (192MB L2) -> gather runs at L2 BW.
// One block per sample; thread d handles dim d; a wave reads 32 consecutive
// dims of one embedding row per step -> 128B coalesced requests.
// ---------------------------------------------------------------------------
__global__ __launch_bounds__(320)
void pool_kernel(const int* __restrict__ ids,
                 const float* __restrict__ emb,
                 float* __restrict__ x) {
  __shared__ int sIds[SEQ];
  const int s = blockIdx.x;
  const int t = threadIdx.x;
  if (t < SEQ) sIds[t] = ids[s * SEQ + t];
  __syncthreads();
  if (t < EMB) {
    float acc = 0.0f;
#pragma unroll 4
    for (int k = 0; k < SEQ; ++k) {
      acc += emb[(size_t)sIds[k] * EMB + t];
    }
    x[(size_t)s * EMB + t] = acc * (1.0f / (float)SEQ);
  }
}

// ---------------------------------------------------------------------------
// Kernel 2: per-aspect MLP on a 64-sample tile using V_WMMA_F32_16X16X4_F32.
// grid = (BATCH/ROWS, NASP); block = 256 threads (8 waves, wave32).
// Wave w computes H columns [w*32, w*32+32). Each B-fragment (W1 slice) is
// loaded once and fed to 4 row-tile accumulators -> WMMA:vload ratio 4:2,
// cutting W1 L2 traffic 4x vs a 16-row tile (315MB total, vs 1.26GB gather).
// Dynamic LDS: sX 64x300 f32 (76.8KB) + sH 64x256 f32 (64KB) = 142KB/WG,
// 2 workgroups per 320KB WGP.
// ---------------------------------------------------------------------------
__global__ __launch_bounds__(256)
void mlp_kernel(const float* __restrict__ x,
                const int* __restrict__ aspect_id,
                const float* __restrict__ W1,
                const float* __restrict__ B1,
                const float* __restrict__ W2,
                const float* __restrict__ B2,
                float* __restrict__ out) {
  extern __shared__ float smem[];
  float* sX = smem;               // ROWS*EMB floats
  float* sH = smem + ROWS * EMB;  // ROWS*HID floats

  const int a    = blockIdx.y;
  const int row0 = blockIdx.x * ROWS;
  const int t    = threadIdx.x;

  // Stage pooled X tile (contiguous) into LDS.
  for (int i = t; i < ROWS * EMB; i += 256) {
    sX[i] = x[(size_t)row0 * EMB + i];
  }
  __syncthreads();

  const int lane  = t & 31;          // wave32
  const int wv    = t >> 5;          // 0..7
  const int nloc  = lane & 15;       // N within 16-wide tile
  const int khalf = (lane >> 4) * 2; // 0 or 2: K-pair selector (ISA A 16x4 layout)
  const int m     = lane & 15;       // A-matrix row held by this lane

  const float* w1a = W1 + (size_t)a * EMB * HID;

  for (int nt = 0; nt < 2; ++nt) {
    const int col = wv * 32 + nt * 16 + nloc;   // global H column, 0..255
    v8f acc[RT];
#pragma unroll
    for (int rt = 0; rt < RT; ++rt) acc[rt] = (v8f){};

    for (int kk = 0; kk < EMB / 4; ++kk) {      // 75 k-steps, 4 WMMAs each
      const int k = kk * 4 + khalf;
      // B-frag: lane holds W1[k][col], W1[k+1][col] (B^T mirrors A layout).
      v2f bfrag;
      bfrag.x = w1a[(size_t)k * HID + col];
      bfrag.y = w1a[(size_t)(k + 1) * HID + col];
#pragma unroll
      for (int rt = 0; rt < RT; ++rt) {
        // A-frag: lane holds X[m][k], X[m][k+1] (8B aligned: k even, pitch 1200B)
        v2f afrag = *(const v2f*)&sX[(rt * 16 + m) * EMB + k];
        acc[rt] = __builtin_amdgcn_wmma_f32_16x16x4_f32(
            /*neg_a=*/false, afrag, /*neg_b=*/false, bfrag,
            /*c_mod=*/(short)0, acc[rt], /*reuse_a=*/false, /*reuse_b=*/false);
      }
    }
    // bias + relu, scatter C/D layout: element r of lane -> row r + 8*(lane>>4)
    const float bias = B1[a * HID + col];
#pragma unroll
    for (int rt = 0; rt < RT; ++rt) {
#pragma unroll
      for (int r = 0; r < 8; ++r) {
        const int rowm = rt * 16 + r + 8 * (lane >> 4);
        float h = acc[rt][r] + bias;
        sH[rowm * HID + col] = h > 0.0f ? h : 0.0f;
      }
    }
  }
  __syncthreads();

  // GEMM2 (256 -> 3) + routed select. 192 active threads, trivial work.
  if (t < ROWS * OUTD) {
    const int s = t / OUTD, j = t % OUTD;
    const float* w2a = W2 + (size_t)a * HID * OUTD;
    float acc = B2[a * OUTD + j];
    for (int k = 0; k < HID; ++k) {
      acc += sH[s * HID + k] * w2a[k * OUTD + j];
    }
    const int b = row0 + s;
    if (aspect_id[b] == a) {
      out[b * OUTD + j] = acc;
    }
  }
}

// ---------------------------------------------------------------------------
extern "C" void kernel_launch(void* const* d_in, const int* in_sizes, int n_in,
                              void* d_out, int out_size, void* d_ws, size_t ws_size,
                              hipStream_t stream) {
  (void)in_sizes; (void)n_in; (void)out_size; (void)ws_size;
  const int*   ids    = (const int*)d_in[0];    // [8192,128] int32
  const int*   aspect = (const int*)d_in[1];    // [8192] int32
  const float* emb    = (const float*)d_in[2];  // [50000,300]
  const float* W1     = (const float*)d_in[3];  // [8,300,256]
  const float* b1     = (const float*)d_in[4];  // [8,256]
  const float* W2     = (const float*)d_in[5];  // [8,256,3]
  const float* b2     = (const float*)d_in[6];  // [8,3]
  float* out = (float*)d_out;                   // [8192,3]
  float* xws = (float*)d_ws;                    // 8192*300 f32 = 9.83 MB scratch

  pool_kernel<<<BATCH, 320, 0, stream>>>(ids, emb, xws);

  const size_t smem_bytes = (size_t)(ROWS * EMB + ROWS * HID) * sizeof(float); // 142336
  dim3 grid(BATCH / ROWS, NASP);
  mlp_kernel<<<grid, 256, smem_bytes, stream>>>(xws, aspect, W1, b1, W2, b2, out);
}